// GatedExternalMemory_26749056319686
// MI455X (gfx1250) — compile-verified
//
#include <hip/hip_runtime.h>
#include <math.h>

#define B_ 16
#define L_ 2048
#define D_ 2048
#define M_ 64
#define S_ 256
#define H_ 4
#define HD_ 64
#define DH_ 512
#define SEG_ 16
#define LSEG_ (L_ / SEG_)   // 128

typedef __attribute__((ext_vector_type(2))) float v2f;
typedef __attribute__((ext_vector_type(8))) float v8f;

// ---------------------------------------------------------------------------
// Phase 1: partial sums of x over L segments, float4-wide.
// grid = B * SEG * (D/1024) = 512 blocks x 256 threads.
// Each block streams LSEG_ rows of a 4KB-wide column slab.
// ---------------------------------------------------------------------------
__global__ void k_partial(const float* __restrict__ x, float* __restrict__ partial) {
    int tid = threadIdx.x;
    int bi  = blockIdx.x;
    int b      = bi >> 5;        // 32 blocks per batch (16 seg * 2 dchunk)
    int rest   = bi & 31;
    int seg    = rest >> 1;
    int dchunk = rest & 1;
    int d = dchunk * 1024 + tid * 4;
    const float4* p = (const float4*)(x + ((size_t)b * L_ + (size_t)seg * LSEG_) * D_ + d);
    float4 acc = make_float4(0.f, 0.f, 0.f, 0.f);
    for (int l = 0; l < LSEG_; ++l) {
        float4 v = p[(size_t)l * (D_ / 4)];
        acc.x += v.x; acc.y += v.y; acc.z += v.z; acc.w += v.w;
    }
    *(float4*)(partial + ((size_t)b * SEG_ + seg) * D_ + d) = acc;
}

// Phase 2: xs = sum(partials)/L, float4-wide.  grid = B*(D/1024) = 32 blocks.
__global__ void k_xs(const float* __restrict__ partial, float* __restrict__ xs) {
    int tid = threadIdx.x;
    int b      = blockIdx.x >> 1;
    int dchunk = blockIdx.x & 1;
    int d = dchunk * 1024 + tid * 4;
    float4 acc = make_float4(0.f, 0.f, 0.f, 0.f);
    for (int s = 0; s < SEG_; ++s) {
        float4 v = *(const float4*)(partial + ((size_t)b * SEG_ + s) * D_ + d);
        acc.x += v.x; acc.y += v.y; acc.z += v.z; acc.w += v.w;
    }
    const float inv = 1.0f / (float)L_;
    acc.x *= inv; acc.y *= inv; acc.z *= inv; acc.w *= inv;
    *(float4*)(xs + (size_t)b * D_ + d) = acc;
}

// ---------------------------------------------------------------------------
// WMMA helper: C(16x16) = A(16xK) * B(Kx16) using V_WMMA_F32_16X16X4_F32.
// A row-major (lda), B row-major (ldb). One wave per tile; EXEC all ones.
// A frag: lane<16 -> row=lane, K={k,k+1}; lane>=16 -> row=lane-16, K={k+2,k+3}.
// B frag mirrors with col = lane%16.
// ---------------------------------------------------------------------------
__device__ __forceinline__ v8f wmma_kloop(const float* __restrict__ A, int lda,
                                          const float* __restrict__ Bm, int ldb,
                                          int K, int lane) {
    int half = lane >> 4;
    int idx  = lane & 15;
    int k0   = half << 1;
    v8f acc = {0.f, 0.f, 0.f, 0.f, 0.f, 0.f, 0.f, 0.f};
    for (int k = 0; k < K; k += 4) {
        v2f a, bb;
        a.x  = A[(size_t)idx * lda + k + k0];
        a.y  = A[(size_t)idx * lda + k + k0 + 1];
        bb.x = Bm[(size_t)(k + k0) * ldb + idx];
        bb.y = Bm[(size_t)(k + k0 + 1) * ldb + idx];
        acc = __builtin_amdgcn_wmma_f32_16x16x4_f32(false, a, false, bb,
                                                    (short)0, acc, false, false);
    }
    return acc;
}

// ---------------------------------------------------------------------------
// Projections off xs: Y[0]=xs@Wq_all, Y[1]=xs@Wv_all, Y[2]=xs@We_all,
// Y[3]=xs@Wsi  (each [16,256], pre-bias).  Block 64 computes glin=xs@Wg^T.
// grid = 65 blocks x 32 threads (1 wave).
// ---------------------------------------------------------------------------
__global__ void k_proj(const float* __restrict__ xs,
                       const float* __restrict__ Wq, const float* __restrict__ Wv,
                       const float* __restrict__ We, const float* __restrict__ Wsi,
                       const float* __restrict__ Wg,
                       float* __restrict__ Y, float* __restrict__ glin) {
    int bi   = blockIdx.x;
    int lane = threadIdx.x;
    if (bi == 64) {  // glin[b][h] = xs[b] . Wg[h]
        for (int o = lane; o < B_ * H_; o += 32) {
            int b = o >> 2, h = o & 3;
            const float* xr = xs + b * D_;
            const float* wr = Wg + (size_t)h * D_;
            float acc = 0.f;
            for (int d = 0; d < D_; ++d) acc += xr[d] * wr[d];
            glin[o] = acc;
        }
        return;
    }
    int w = bi >> 4;        // weight family
    int t = bi & 15;        // n-tile
    int nbase = t * 16;
    const float* Bp; int ldb;
    if (w == 0)      { int h = nbase >> 6; Bp = Wq  + (size_t)h * D_ * HD_ + (nbase - h * 64); ldb = HD_; }
    else if (w == 1) { int h = nbase >> 6; Bp = Wv  + (size_t)h * D_ * HD_ + (nbase - h * 64); ldb = HD_; }
    else if (w == 2) { int h = nbase >> 6; Bp = We  + (size_t)h * D_ * M_  + (nbase - h * 64); ldb = M_; }
    else             {                     Bp = Wsi + nbase;                                   ldb = S_; }
    v8f acc = wmma_kloop(xs, D_, Bp, ldb, D_, lane);
    int half = lane >> 4, idx = lane & 15;
    float* Yw = Y + (size_t)w * (B_ * S_);
    #pragma unroll
    for (int j = 0; j < 8; ++j) {
        int row = j + 8 * half;  // batch index
        Yw[row * S_ + nbase + idx] = acc[j];
    }
}

// ---------------------------------------------------------------------------
// Attention + write path, one block per batch.  Avoids materializing keys:
// scores[h,m] = mem[b,m,h*64:] . (Wk[h] q[h]) + q[h].bk[h]
// ---------------------------------------------------------------------------
__global__ void k_small(const float* __restrict__ memory,
                        const float* __restrict__ Y, const float* __restrict__ glin,
                        const float* __restrict__ bq,
                        const float* __restrict__ Wk, const float* __restrict__ bk,
                        const float* __restrict__ Wp, const float* __restrict__ bp,
                        const float* __restrict__ bg, const float* __restrict__ bv,
                        const float* __restrict__ be, const float* __restrict__ bsi,
                        const float* __restrict__ Wsp, const float* __restrict__ bsp,
                        const float* __restrict__ thr,
                        float* __restrict__ ph_out, float* __restrict__ mem_out) {
    int b = blockIdx.x;
    int tid = threadIdx.x;           // 256 threads
    __shared__ float q[H_ * HD_];
    __shared__ float qk[H_ * HD_];
    __shared__ float qb[H_];
    __shared__ float sc[H_ * M_];
    __shared__ float r[H_ * HD_];
    __shared__ float wc_s[S_];
    __shared__ float er[H_ * M_];
    __shared__ float mm[S_];
    __shared__ float upd[S_];
    __shared__ float te[M_];
    __shared__ float red[256];
    __shared__ float wg[H_];
    __shared__ float sgate;

    const float* memb = memory + (size_t)b * M_ * S_;
    const float* Y1 = Y + 0 * (B_ * S_) + b * S_;
    const float* Y2 = Y + 1 * (B_ * S_) + b * S_;
    const float* Y3 = Y + 2 * (B_ * S_) + b * S_;
    const float* Y4 = Y + 3 * (B_ * S_) + b * S_;

    int h = tid >> 6, kk = tid & 63;          // tid = h*64 + (k|m)
    q[tid]    = Y1[tid] + bq[h * HD_ + kk];
    wc_s[tid] = Y2[tid] + bv[h * HD_ + kk];
    er[tid]   = 1.f / (1.f + expf(-(Y3[tid] + be[h * M_ + kk])));
    __syncthreads();

    {   // qk[h][kk] = sum_j Wk[h][kk][j] * q[h][j]
        const float* wr = Wk + ((size_t)h * HD_ + kk) * HD_;
        const float* qh = q + h * HD_;
        float acc = 0.f;
        for (int j = 0; j < HD_; ++j) acc += wr[j] * qh[j];
        qk[tid] = acc;
    }
    if (tid < H_) {
        float acc = 0.f;
        for (int j = 0; j < HD_; ++j) acc += q[tid * HD_ + j] * bk[tid * HD_ + j];
        qb[tid] = acc;
    }
    __syncthreads();

    {   // scores[h][m]
        const float* mr  = memb + kk * S_ + h * HD_;   // kk plays role of m
        const float* qkh = qk + h * HD_;
        float acc = 0.f;
        for (int j = 0; j < HD_; ++j) acc += mr[j] * qkh[j];
        sc[tid] = (acc + qb[h]) * 0.125f;              // /sqrt(64)
    }
    __syncthreads();

    if (tid < H_) {  // softmax over m=64, one head per thread
        float mx = -1e30f;
        for (int m = 0; m < M_; ++m) mx = fmaxf(mx, sc[tid * M_ + m]);
        float s = 0.f;
        for (int m = 0; m < M_; ++m) { float e = expf(sc[tid * M_ + m] - mx); sc[tid * M_ + m] = e; s += e; }
        float inv = 1.f / s;
        for (int m = 0; m < M_; ++m) sc[tid * M_ + m] *= inv;
    }
    __syncthreads();

    {   // r[h][k] = sum_m attn[h][m] * mem[b][m][h*64+k]
        float acc = 0.f;
        for (int m = 0; m < M_; ++m) acc += sc[h * M_ + m] * memb[m * S_ + h * HD_ + kk];
        r[tid] = acc;
    }
    __syncthreads();

    // ph[b][h*512+p] = r[h] . Wp[h][:,p] + bp
    for (int o = tid; o < D_; o += 256) {
        int hh = o >> 9, p = o & 511;
        const float* rh = r + hh * HD_;
        const float* wp = Wp + (size_t)hh * HD_ * DH_ + p;
        float acc = bp[hh * DH_ + p];
        for (int k = 0; k < HD_; ++k) acc += rh[k] * wp[(size_t)k * DH_];
        ph_out[(size_t)b * D_ + o] = acc;
    }

    // ---- write path ----
    if (tid < H_) wg[tid] = 1.f / (1.f + expf(-(glin[b * H_ + tid] + bg[tid])));
    {   // memmean[s]
        float acc = 0.f;
        for (int m = 0; m < M_; ++m) acc += memb[m * S_ + tid];
        mm[tid] = acc * (1.0f / (float)M_);
    }
    __syncthreads();
    {   // surprise
        float xp = Y4[tid] + bsi[tid];
        float pr = bsp[tid];
        for (int t2 = 0; t2 < S_; ++t2) pr += mm[t2] * Wsp[t2 * S_ + tid];
        float dlt = xp - pr;
        red[tid] = dlt * dlt;
    }
    __syncthreads();
    for (int off = 128; off > 0; off >>= 1) {
        if (tid < off) red[tid] += red[tid + off];
        __syncthreads();
    }
    if (tid == 0) {
        float surprise = red[0] * (1.0f / (float)S_);
        sgate = 1.f / (1.f + expf(-(surprise - thr[0])));
    }
    __syncthreads();
    upd[tid] = wg[h] * sgate * wc_s[tid];
    if (tid < M_) {
        float mx = er[tid];
        mx = fmaxf(mx, er[M_ + tid]);
        mx = fmaxf(mx, er[2 * M_ + tid]);
        mx = fmaxf(mx, er[3 * M_ + tid]);
        te[tid] = mx;
    }
    __syncthreads();
    for (int m = 0; m < M_; ++m)
        mem_out[(size_t)b * M_ * S_ + m * S_ + tid] =
            memb[m * S_ + tid] * (1.f - te[m]) + upd[tid];
}

// ---------------------------------------------------------------------------
// read_out = ph @ Wc + bc   ([16,2048]x[2048,2048]) — WMMA, 128 n-tiles.
// ---------------------------------------------------------------------------
__global__ void k_readout(const float* __restrict__ ph, const float* __restrict__ Wc,
                          const float* __restrict__ bc, float* __restrict__ ro) {
    int t = blockIdx.x;
    int lane = threadIdx.x;
    int nbase = t * 16;
    v8f acc = wmma_kloop(ph, D_, Wc + nbase, D_, D_, lane);
    int half = lane >> 4, idx = lane & 15;
    float bias = bc[nbase + idx];
    #pragma unroll
    for (int j = 0; j < 8; ++j) {
        int row = j + 8 * half;
        ro[(size_t)row * D_ + nbase + idx] = acc[j] + bias;
    }
}

// ---------------------------------------------------------------------------
// x_aug = x + read_out broadcast; float4 streaming.  grid=65536 x 256.
// ---------------------------------------------------------------------------
__global__ void k_aug(const float* __restrict__ x, const float* __restrict__ ro,
                      float* __restrict__ out) {
    size_t i = (size_t)blockIdx.x * blockDim.x + threadIdx.x;   // float4 index
    size_t e = i * 4;
    int b = (int)(e / ((size_t)L_ * D_));
    int d = (int)(e % D_);
    float4 xv = ((const float4*)x)[i];
    const float4 rv = *(const float4*)(ro + (size_t)b * D_ + d);
    xv.x += rv.x; xv.y += rv.y; xv.z += rv.z; xv.w += rv.w;
    ((float4*)out)[i] = xv;
}

extern "C" void kernel_launch(void* const* d_in, const int* in_sizes, int n_in,
                              void* d_out, int out_size, void* d_ws, size_t ws_size,
                              hipStream_t stream) {
    (void)in_sizes; (void)n_in; (void)out_size; (void)ws_size;
    const float* x      = (const float*)d_in[0];
    const float* memory = (const float*)d_in[1];
    const float* Wq  = (const float*)d_in[2];  const float* bq  = (const float*)d_in[3];
    const float* Wk  = (const float*)d_in[4];  const float* bk  = (const float*)d_in[5];
    const float* Wp  = (const float*)d_in[6];  const float* bp  = (const float*)d_in[7];
    const float* Wg  = (const float*)d_in[8];  const float* bg  = (const float*)d_in[9];
    const float* Wv  = (const float*)d_in[10]; const float* bv  = (const float*)d_in[11];
    const float* We  = (const float*)d_in[12]; const float* be  = (const float*)d_in[13];
    const float* Wc  = (const float*)d_in[14]; const float* bc  = (const float*)d_in[15];
    const float* Wsi = (const float*)d_in[16]; const float* bsi = (const float*)d_in[17];
    const float* Wsp = (const float*)d_in[18]; const float* bsp = (const float*)d_in[19];
    const float* thr = (const float*)d_in[20];

    float* out     = (float*)d_out;
    float* mem_out = out + (size_t)B_ * L_ * D_;

    float* ws      = (float*)d_ws;
    float* partial = ws;                          // B*SEG*D   = 524288
    float* xs      = partial + B_ * SEG_ * D_;    // B*D       = 32768
    float* Y       = xs + B_ * D_;                // 4*B*S     = 16384
    float* glin    = Y + 4 * B_ * S_;             // B*H       = 64
    float* ph      = glin + B_ * H_;              // B*D       = 32768
    float* ro      = ph + B_ * D_;                // B*D       = 32768

    k_partial<<<B_ * SEG_ * (D_ / 1024), 256, 0, stream>>>(x, partial);
    k_xs<<<B_ * (D_ / 1024), 256, 0, stream>>>(partial, xs);
    k_proj<<<65, 32, 0, stream>>>(xs, Wq, Wv, We, Wsi, Wg, Y, glin);
    k_small<<<B_, 256, 0, stream>>>(memory, Y, glin, bq, Wk, bk, Wp, bp,
                                    bg, bv, be, bsi, Wsp, bsp, thr, ph, mem_out);
    k_readout<<<D_ / 16, 32, 0, stream>>>(ph, Wc, bc, ro);
    k_aug<<<(int)(((size_t)B_ * L_ * D_ / 4) / 256), 256, 0, stream>>>(x, ro, out);
}